// DiscreteNormalization_88776974008602
// MI455X (gfx1250) — compile-verified
//
#include <hip/hip_runtime.h>
#include <hip/hip_bf16.h>
#include <stdint.h>

// WiSARD weightless NN: S=8 subnets, N=8192 neurons, B=12 bits/neuron, IB=8192 input bits.
#define S_SUB   8
#define N_NEU   8192
#define B_BITS  12
#define IB_BITS 8192
#define TBL     (1 << B_BITS)        // 4096 cells per neuron
#define PACK_WORDS (IB_BITS / 32)    // 256 dwords = 1 KB packed input
#define NPB     32                   // neurons per block (block = 32 n x 8 s = 256 thr)

typedef __attribute__((ext_vector_type(4))) int v4i;

// ---- CDNA5 async global->LDS DMA availability -------------------------------
#if defined(__has_builtin)
#  if __has_builtin(__builtin_amdgcn_global_load_async_to_lds_b32)
#    define HAVE_ASYNC_LDS 1
#  else
#    define HAVE_ASYNC_LDS 0
#  endif
#else
#  define HAVE_ASYNC_LDS 0
#endif

// Builtin takes (AS1 int* src, AS3 int* lds_dst, imm offset, imm cpol).
// Route casts through integers: generic LDS pointers carry the AS3 offset in
// their low 32 bits; generic global pointers are bit-identical to AS1.
#define AS1_I32(p) ((__attribute__((address_space(1))) int*)(uintptr_t)(p))
#define AS3_I32(p) ((__attribute__((address_space(3))) int*)(uint32_t)(uintptr_t)(p))

// -----------------------------------------------------------------------------
// Kernel 1: bit-pack x[8192] (int32 0/1) into 256 dwords in d_ws.
// Single 256-thread block; each thread packs 32 consecutive input bits.
// -----------------------------------------------------------------------------
__global__ __launch_bounds__(PACK_WORDS) void wnn_pack_x(
    const int* __restrict__ x, uint32_t* __restrict__ packed) {
  const int t = threadIdx.x;  // 0..255
  const v4i* x4 = (const v4i*)x;
  uint32_t w = 0;
#pragma unroll
  for (int j = 0; j < 8; ++j) {
    v4i v = __builtin_nontemporal_load(x4 + t * 8 + j);  // x[t*32+4j .. +3]
    w |= (uint32_t)(v.x & 1) << (4 * j + 0);
    w |= (uint32_t)(v.y & 1) << (4 * j + 1);
    w |= (uint32_t)(v.z & 1) << (4 * j + 2);
    w |= (uint32_t)(v.w & 1) << (4 * j + 3);
  }
  packed[t] = w;
}

__device__ __forceinline__ uint32_t xbit(const uint32_t* sx, int idx) {
  // idx guaranteed in [0, 8192) by construction of conn
  return (sx[idx >> 5] >> (idx & 31)) & 1u;
}

// -----------------------------------------------------------------------------
// Kernel 2: one thread per (subnet, neuron) pair -> 65536 threads in 256 blocks,
// spreading the 64K independent random RAM-table gathers across the whole chip
// (this kernel is pure memory-latency bound; MLP breadth is everything).
//   thread t: s = t>>5 (wave index), nl = t&31 (lane), n = blockIdx*32 + nl.
// Packed x (1 KB) is DMA'd into LDS via the CDNA5 async global->LDS path.
// Vote reduction across the 8 s-waves uses an LDS ds_add atomic.
// -----------------------------------------------------------------------------
__global__ __launch_bounds__(256) void wnn_main(
    const uint32_t* __restrict__ packedx,   // 256 dwords in d_ws
    const int* __restrict__ conn,           // [S][N][B]
    const float* __restrict__ memory,       // [S][N][4096]
    float* __restrict__ out) {              // [N]
  __shared__ uint32_t sx[PACK_WORDS];
  __shared__ int votes[NPB];

  const int t  = threadIdx.x;   // 0..255
  const int s  = t >> 5;        // 0..7  (one wave32 per subnet)
  const int nl = t & 31;        // 0..31 (lane -> neuron within tile)
  const int n  = blockIdx.x * NPB + nl;

  if (t < NPB) votes[t] = 0;

#if HAVE_ASYNC_LDS
  // One async 32-bit DMA word per lane: 1 KB broadcast table -> LDS,
  // tracked by ASYNCcnt (CDNA5 async global->LDS path).
  __builtin_amdgcn_global_load_async_to_lds_b32(
      AS1_I32(packedx + t), AS3_I32(&sx[t]), 0, 0);
#  if __has_builtin(__builtin_amdgcn_s_wait_asynccnt)
  __builtin_amdgcn_s_wait_asynccnt(0);
#  else
  asm volatile("s_wait_asynccnt 0" ::: "memory");
#  endif
#else
  sx[t] = packedx[t];
#endif
  __syncthreads();

  // conn row: 12 consecutive ints, 48 B, 16 B aligned -> 3x b128 NT loads.
  const v4i* c = (const v4i*)(conn + ((size_t)s * N_NEU + n) * B_BITS);
  v4i c0 = __builtin_nontemporal_load(c + 0);
  v4i c1 = __builtin_nontemporal_load(c + 1);
  v4i c2 = __builtin_nontemporal_load(c + 2);

  uint32_t addr =
      (xbit(sx, c0.x) << 0)  | (xbit(sx, c0.y) << 1)  |
      (xbit(sx, c0.z) << 2)  | (xbit(sx, c0.w) << 3)  |
      (xbit(sx, c1.x) << 4)  | (xbit(sx, c1.y) << 5)  |
      (xbit(sx, c1.z) << 6)  | (xbit(sx, c1.w) << 7)  |
      (xbit(sx, c2.x) << 8)  | (xbit(sx, c2.y) << 9)  |
      (xbit(sx, c2.z) << 10) | (xbit(sx, c2.w) << 11);

  // Random 4 B gather from the 1 GB RAM table: NT so it doesn't thrash L2.
  float v = __builtin_nontemporal_load(
      memory + ((size_t)s * N_NEU + n) * (size_t)TBL + addr);

  // Cross-wave majority reduction: one ds_add_u32 per (s,n) thread.
  atomicAdd(&votes[nl], (v != 0.0f) ? 1 : 0);
  __syncthreads();

  if (t < NPB) {
    out[blockIdx.x * NPB + t] = (votes[t] > (S_SUB / 2)) ? 1.0f : 0.0f;
  }
}

// -----------------------------------------------------------------------------
extern "C" void kernel_launch(void* const* d_in, const int* in_sizes, int n_in,
                              void* d_out, int out_size, void* d_ws, size_t ws_size,
                              hipStream_t stream) {
  (void)in_sizes; (void)n_in; (void)out_size; (void)ws_size;
  const int*   x      = (const int*)d_in[0];     // [8192] int32 0/1
  const int*   conn   = (const int*)d_in[1];     // [8][8192][12] int32
  const float* memory = (const float*)d_in[2];   // [8][8192][4096] f32 0/1
  float*       out    = (float*)d_out;           // [8192] f32
  uint32_t*    packed = (uint32_t*)d_ws;         // 1 KB scratch

  wnn_pack_x<<<1, PACK_WORDS, 0, stream>>>(x, packed);
  wnn_main<<<N_NEU / NPB, 256, 0, stream>>>(packed, conn, memory, out);
}